// NegNCE_24721831755835
// MI455X (gfx1250) — compile-verified
//
#include <hip/hip_runtime.h>
#include <cstdint>

// ---------------------------------------------------------------------------
// NegNCE loss, N = 8192, one workgroup per row.
// One HBM pass over sim (256 MB ~ 11 us @ 23.3 TB/s). Row + diag + exp cache
// staged in LDS (96 KB of the WGP's 320 KB) via CDNA5 async global->LDS
// copies (ASYNCcnt / s_wait_asynccnt). All softmax math in base-2 domain so
// exp/log are single v_exp_f32 / v_log_f32 TRANS ops (co-execute with VALU).
// ---------------------------------------------------------------------------

#define NN        8192
#define TPB       256              // 8 wave32s
#define WAVES     (TPB / 32)
#define VEC_ITERS (NN / (TPB * 4)) // 8 float4s per thread

static constexpr float kLog2e100 = 144.26950408889634f;  // 100 * log2(e)
static constexpr float kLn2      = 0.6931471805599453f;
static constexpr float kEps      = 1e-6f;
static constexpr float kNegW     = 0.5f;

struct Accum {
    float    pos_sum;   // sum_i -log(clip(p_ii))
    float    neg_sum;   // sum over selected -log(1 - clip(p_ij))
    unsigned cnt;       // number of selected hard negatives
    unsigned pad;
};

// --------------------------- kernel 1: init --------------------------------
__global__ __launch_bounds__(TPB) void negnce_init_kernel(
    const float* __restrict__ sim, float* __restrict__ diag, Accum* acc) {
    int i = blockIdx.x * TPB + threadIdx.x;
    if (i < NN) diag[i] = sim[(size_t)i * NN + i];
    if (i == 0) { acc->pos_sum = 0.f; acc->neg_sum = 0.f; acc->cnt = 0u; acc->pad = 0u; }
}

// --------------------------- kernel 2: per-row -----------------------------
__global__ __launch_bounds__(TPB) void negnce_row_kernel(
    const float* __restrict__ sim, const float* __restrict__ diag, Accum* acc) {
    __shared__ __align__(16) float4 sx4[NN / 4];   // row of sim
    __shared__ __align__(16) float4 se4[NN / 4];   // 2^(K*(x - rowmax)) cache
    __shared__ __align__(16) float4 sd4[NN / 4];   // full diagonal vector
    __shared__ float    redf[WAVES];
    __shared__ unsigned redu[WAVES];

    const int tid  = threadIdx.x;
    const int row  = blockIdx.x;
    const int lane = tid & 31;
    const int wid  = tid >> 5;

    // ---- stage row + diagonal into LDS with async b128 copies (ASYNCcnt) ----
    {
        const unsigned sx_base = (unsigned)(uintptr_t)&sx4[0];
        const unsigned sd_base = (unsigned)(uintptr_t)&sd4[0];
        const float* rowp = sim + (size_t)row * NN;
#pragma unroll
        for (int k = 0; k < VEC_ITERS; ++k) {
            const unsigned fidx  = (unsigned)(k * TPB + tid) * 4u;  // float index
            const unsigned loffx = sx_base + fidx * 4u;             // LDS byte addr
            const uint64_t gax   = (uint64_t)(uintptr_t)(rowp + fidx);
            asm volatile("global_load_async_to_lds_b128 %0, %1, off"
                         :: "v"(loffx), "v"(gax) : "memory");
            const unsigned loffd = sd_base + fidx * 4u;
            const uint64_t gad   = (uint64_t)(uintptr_t)(diag + fidx);
            asm volatile("global_load_async_to_lds_b128 %0, %1, off"
                         :: "v"(loffd), "v"(gad) : "memory");
        }
        asm volatile("s_wait_asynccnt 0" ::: "memory");
    }
    __syncthreads();

    // ---- pass A: row max (b128 LDS reads, lane-contiguous) ----
    float m = -3.402823466e+38f;
#pragma unroll
    for (int k = 0; k < VEC_ITERS; ++k) {
        const float4 v = sx4[k * TPB + tid];
        m = fmaxf(m, fmaxf(fmaxf(v.x, v.y), fmaxf(v.z, v.w)));
    }
#pragma unroll
    for (int o = 16; o > 0; o >>= 1) m = fmaxf(m, __shfl_xor(m, o, 32));
    if (lane == 0) redf[wid] = m;
    __syncthreads();
    if (tid < 32) {
        float v = (lane < WAVES) ? redf[lane] : -3.402823466e+38f;
#pragma unroll
        for (int o = WAVES / 2; o > 0; o >>= 1) v = fmaxf(v, __shfl_xor(v, o, 32));
        if (lane == 0) redf[0] = v;
    }
    __syncthreads();
    const float negKm = -kLog2e100 * redf[0];   // -K * rowmax
    __syncthreads();

    // ---- pass B: e = 2^(K*x - K*max)  (1 FMA + 1 v_exp_f32 per element) ----
    float ssum = 0.f;
#pragma unroll
    for (int k = 0; k < VEC_ITERS; ++k) {
        const int i = k * TPB + tid;
        const float4 v = sx4[i];
        float4 e;
        e.x = __builtin_amdgcn_exp2f(fmaf(kLog2e100, v.x, negKm));
        e.y = __builtin_amdgcn_exp2f(fmaf(kLog2e100, v.y, negKm));
        e.z = __builtin_amdgcn_exp2f(fmaf(kLog2e100, v.z, negKm));
        e.w = __builtin_amdgcn_exp2f(fmaf(kLog2e100, v.w, negKm));
        se4[i] = e;
        ssum += (e.x + e.y) + (e.z + e.w);
    }
#pragma unroll
    for (int o = 16; o > 0; o >>= 1) ssum += __shfl_xor(ssum, o, 32);
    if (lane == 0) redf[wid] = ssum;
    __syncthreads();
    if (tid < 32) {
        float v = (lane < WAVES) ? redf[lane] : 0.f;
#pragma unroll
        for (int o = WAVES / 2; o > 0; o >>= 1) v += __shfl_xor(v, o, 32);
        if (lane == 0) redf[0] = v;
    }
    __syncthreads();
    const float rcpZ = 1.0f / redf[0];
    __syncthreads();

    // ---- pass C: hard-negative terms, log2 domain ----
    // MARGIN==0: selected <=> x > min(d_i, d_j). Diagonal self-excludes
    // (x_ii == d_i, strict compare), matching the reference's eye mask.
    const float d_i = ((const float*)sd4)[row];
    float neg_l2 = 0.f;          // sum of log2(1 - p) over selected
    unsigned scnt = 0;           // wave-uniform count via ballot/popcount (SALU)

    auto proc = [&](float xc, float ec, float djc) {
        float p = ec * rcpZ;
        p = fminf(fmaxf(p, kEps), 1.0f - kEps);
        const bool sel = xc > fminf(d_i, djc);
        const float lg = __builtin_amdgcn_logf(1.0f - p);   // v_log_f32 (log2)
        neg_l2 += sel ? lg : 0.0f;
        scnt += (unsigned)__builtin_popcount(__builtin_amdgcn_ballot_w32(sel));
    };
#pragma unroll
    for (int k = 0; k < VEC_ITERS; ++k) {
        const int i = k * TPB + tid;
        const float4 x = sx4[i];
        const float4 e = se4[i];
        const float4 dj = sd4[i];
        proc(x.x, e.x, dj.x);
        proc(x.y, e.y, dj.y);
        proc(x.z, e.z, dj.z);
        proc(x.w, e.w, dj.w);
    }
#pragma unroll
    for (int o = 16; o > 0; o >>= 1) neg_l2 += __shfl_xor(neg_l2, o, 32);
    if (lane == 0) { redf[wid] = neg_l2; redu[wid] = scnt; }
    __syncthreads();
    if (tid == 0) {
        float tn = 0.f; unsigned tc = 0;
#pragma unroll
        for (int w = 0; w < WAVES; ++w) { tn += redf[w]; tc += redu[w]; }
        // positive term for this row: -ln(clip(p_ii))
        float pii = ((const float*)se4)[row] * rcpZ;
        pii = fminf(fmaxf(pii, kEps), 1.0f - kEps);
        const float posl = -kLn2 * __builtin_amdgcn_logf(pii);
        unsafeAtomicAdd(&acc->neg_sum, -kLn2 * tn);   // -> global_atomic_add_f32
        unsafeAtomicAdd(&acc->pos_sum, posl);
        atomicAdd(&acc->cnt, tc);
    }
}

// --------------------------- kernel 3: finalize ----------------------------
__global__ void negnce_finalize_kernel(const Accum* __restrict__ acc,
                                       float* __restrict__ out) {
    if (threadIdx.x == 0 && blockIdx.x == 0) {
        float loss = acc->pos_sum * (1.0f / (float)NN);   // C_POS_W = 1
        const unsigned c = acc->cnt;
        if (c > 0u) loss += kNegW * acc->neg_sum / (float)c;
        out[0] = loss;
    }
}

// ---------------------------------------------------------------------------
extern "C" void kernel_launch(void* const* d_in, const int* in_sizes, int n_in,
                              void* d_out, int out_size, void* d_ws, size_t ws_size,
                              hipStream_t stream) {
    (void)in_sizes; (void)n_in; (void)out_size; (void)ws_size;
    const float* sim  = (const float*)d_in[0];
    float*       out  = (float*)d_out;
    float*       diag = (float*)d_ws;                         // NN floats
    Accum*       acc  = (Accum*)((char*)d_ws + NN * sizeof(float));

    negnce_init_kernel<<<NN / TPB, TPB, 0, stream>>>(sim, diag, acc);
    negnce_row_kernel <<<NN,       TPB, 0, stream>>>(sim, diag, acc);
    negnce_finalize_kernel<<<1, 64, 0, stream>>>(acc, out);
}